// VectorQuantizer_83631603188439
// MI455X (gfx1250) — compile-verified
//
#include <hip/hip_runtime.h>
#include <hip/hip_bf16.h>
#include <stdint.h>

// ---------------------------------------------------------------------------
// VQ-VAE vector quantizer for MI455X (gfx1250, wave32).
//   B=32, H=64, W=64, C=64, K_emb=1024, D=64 (flat rows run along W per the
//   reference's NCHW reshape; each (b,h) owns a contiguous 16KB z_e chunk).
// Out layout (all float32): [z_q_st NHWC: 8388608][indices: 131072][loss: 1]
// Workspace: ws[0..1023] = -0.5*||emb_k||^2 ; ws[1024..1025] = loss accum.
// ---------------------------------------------------------------------------

typedef __attribute__((ext_vector_type(2))) float v2f;
typedef __attribute__((ext_vector_type(8))) float v8f;

// Only the device pass knows amdgcn builtins; the host pass reports
// __has_builtin == 0 for aux-target builtins, so gate the check.
#if defined(__HIP_DEVICE_COMPILE__) && defined(__gfx1250__)
#if !__has_builtin(__builtin_amdgcn_wmma_f32_16x16x4_f32)
#error "gfx1250 f32 WMMA builtin __builtin_amdgcn_wmma_f32_16x16x4_f32 not available"
#endif
#endif

static __device__ __forceinline__ v8f wmma_f32_4(v2f a, v2f b, v8f c) {
  // D = A(16x4 f32) x B(4x16 f32) + C(16x16 f32)
  return __builtin_amdgcn_wmma_f32_16x16x4_f32(
      /*neg_a=*/false, a, /*neg_b=*/false, b,
      /*c_mod=*/(short)0, c, /*reuse_a=*/false, /*reuse_b=*/false);
}

#define N_TOTAL   8388608      // 32*64*64*64
#define N_ROWS    131072       // 32*64*64
#define IDX_OFF   N_TOTAL
#define LOSS_OFF  (N_TOTAL + N_ROWS)
#define EMB_PAD   68           // floats per LDS emb row (272B, 16B aligned, bank stride 4)
#define CHUNK_F   (64 * EMB_PAD)
#define CHUNK_B   (CHUNK_F * 4)

// --- pre-pass: eh[k] = -0.5 * ||emb_k||^2 ; zero the loss accumulators -------
__global__ __launch_bounds__(256) void vq_e2_kernel(const float* __restrict__ emb,
                                                    float* __restrict__ ws) {
  int k = blockIdx.x * 256 + threadIdx.x;      // grid: 4 x 256 = 1024
  if (k == 0) { ws[1024] = 0.0f; ws[1025] = 0.0f; }
  float s = 0.0f;
  const float4* e4 = (const float4*)(emb + (size_t)k * 64);
#pragma unroll
  for (int i = 0; i < 16; ++i) {
    float4 v = e4[i];
    s += v.x * v.x + v.y * v.y + v.z * v.z + v.w * v.w;
  }
  ws[k] = -0.5f * s;
}

// --- main kernel: one block per (b,h) ---------------------------------------
__global__ __launch_bounds__(256) void vq_main_kernel(const float* __restrict__ z,
                                                      const float* __restrict__ emb,
                                                      float* __restrict__ ws,
                                                      float* __restrict__ out) {
  __shared__ float ehlds[1024];           // -0.5*||e||^2
  __shared__ float embbuf[2 * CHUNK_F];   // double-buffered emb chunks (padded)
  __shared__ float qlds[64 * 65];         // gathered embeddings, stride 65 (bank-safe)
  __shared__ float redval[8 * 16];
  __shared__ int   redidx[8 * 16];
  __shared__ int   fidx[64];

  const int t    = threadIdx.x;
  const int bh   = blockIdx.x;            // b*64 + h
  const int b    = bh >> 6;
  const int h    = bh & 63;
  const float* zb = z + (size_t)bh * 4096;

  // stage eh table (1024 floats) into LDS
  ((float4*)ehlds)[t] = ((const float4*)ws)[t];

  const int lane = t & 31;
  const int wv   = t >> 5;
  const int nt   = wv & 3;                // flat-row tile: rows c0..c0+15
  const int half = wv >> 2;               // which pair of M-subtiles per chunk
  const int ln   = lane & 15;
  const int hsel = lane >> 4;             // 0: K+0/1 or M+0..7 ; 1: K+2/3 or M+8..15
  const int c0   = nt * 16;

  const uint32_t embbuf_base = (uint32_t)(size_t)(&embbuf[0]);

  // async global->LDS stage of one 64-embedding chunk (16KB): 4x B128 per thread
  auto issue_chunk = [&](int ch, int buf) {
#pragma unroll
    for (int jj = 0; jj < 4; ++jj) {
      int li  = t + 256 * jj;             // 0..1023
      int m   = li >> 4;                  // emb row in chunk
      int seg = li & 15;                  // 16B segment in row
      uint32_t ldsb = embbuf_base + (uint32_t)(buf * CHUNK_B + m * (EMB_PAD * 4) + seg * 16);
      uint32_t goff = (uint32_t)(((ch * 64 + m) * 64) * 4 + seg * 16);
      asm volatile("global_load_async_to_lds_b128 %0, %1, %2 offset:0"
                   :: "v"(ldsb), "v"(goff), "s"(emb) : "memory");
    }
  };
  issue_chunk(0, 0);
  issue_chunk(1, 1);

  // B fragments (flat z tile, K=64 x N=16) -> 32 VGPRs, loaded once, coalesced
  v2f B[16];
#pragma unroll
  for (int k = 0; k < 16; ++k) {
    int kk = 4 * k + 2 * hsel;
    B[k].x = zb[kk * 64 + c0 + ln];
    B[k].y = zb[(kk + 1) * 64 + c0 + ln];
  }

  // running argmax of score = dot(x,e) - 0.5*||e||^2  (== argmin distance)
  float runv[8];
  int   runi[8];
#pragma unroll
  for (int g = 0; g < 8; ++g) { runv[g] = -3.0e38f; runi[g] = 0x7fffffff; }

  for (int ch = 0; ch < 16; ++ch) {
    const int buf = ch & 1;
    if (ch < 15) asm volatile("s_wait_asynccnt 0x4" ::: "memory");
    else         asm volatile("s_wait_asynccnt 0x0" ::: "memory");
    __syncthreads();                      // chunk ch resident for all waves

    const float* eb = &embbuf[buf * CHUNK_F];
#pragma unroll
    for (int s2 = 0; s2 < 2; ++s2) {
      const int sub = half * 2 + s2;      // M-subtile 0..3 within chunk
      const int m0  = ch * 64 + sub * 16; // absolute embedding base

      // Preload ALL A-fragments for this subtile, then PIN them in VGPRs with
      // an empty asm so the scheduler cannot sink each ds_load to just before
      // its WMMA (which would cost one s_wait_dscnt round-trip per WMMA pair).
      v2f afrag[16];
      const float* arow = &eb[(sub * 16 + ln) * EMB_PAD + 2 * hsel];
#pragma unroll
      for (int k = 0; k < 16; ++k) {
        afrag[k] = *(const v2f*)&arow[4 * k];
      }
#pragma unroll
      for (int k = 0; k < 16; ++k) {
        asm volatile("" : "+v"(afrag[k]));
      }

      v8f acc;
#pragma unroll
      for (int g = 0; g < 8; ++g) acc[g] = ehlds[m0 + g + 8 * hsel]; // broadcast reads
#pragma unroll
      for (int k = 0; k < 16; ++k) {
        // A fragment: lane L<16 -> emb row m0+L, K = 4k..4k+1 ; lanes 16-31 -> K+2/3
        acc = wmma_f32_4(afrag[k], B[k], acc);
      }
#pragma unroll
      for (int g = 0; g < 8; ++g) {
        int  cand   = m0 + g + 8 * hsel;
        bool better = acc[g] > runv[g];   // strict > keeps first (lowest) index
        runv[g] = better ? acc[g] : runv[g];
        runi[g] = better ? cand   : runi[g];
      }
    }
    __syncthreads();                      // everyone done reading buf
    if (ch + 2 < 16) issue_chunk(ch + 2, buf);
  }

  // in-wave reduce across the 8 accumulator rows (tie-break: smaller index)
  float bv = runv[0];
  int   bi = runi[0];
#pragma unroll
  for (int g = 1; g < 8; ++g) {
    bool better = (runv[g] > bv) || (runv[g] == bv && runi[g] < bi);
    bv = better ? runv[g] : bv;
    bi = better ? runi[g] : bi;
  }
  // lanes L and L+16 hold the same flat row (M split 0..7 / 8..15): merge
  {
    float ov = __shfl_xor(bv, 16, 32);
    int   oi = __shfl_xor(bi, 16, 32);
    bool better = (ov > bv) || (ov == bv && oi < bi);
    bv = better ? ov : bv;
    bi = better ? oi : bi;
  }
  if (lane < 16) { redval[wv * 16 + lane] = bv; redidx[wv * 16 + lane] = bi; }
  __syncthreads();

  // merge the two waves covering each row tile; emit indices (as f32 values)
  if (t < 64) {
    int c   = t;
    int nt2 = c >> 4, r = c & 15;
    float v0 = redval[nt2 * 16 + r];       int i0 = redidx[nt2 * 16 + r];
    float v1 = redval[(4 + nt2) * 16 + r]; int i1 = redidx[(4 + nt2) * 16 + r];
    bool better = (v1 > v0) || (v1 == v0 && i1 < i0);
    int bi2 = better ? i1 : i0;
    fidx[c] = bi2;
    out[IDX_OFF + (size_t)b * 4096 + c * 64 + h] = (float)bi2;  // flat n=(b*64+c)*64+h
  }
  __syncthreads();

  // gather chosen embeddings (coalesced per row) into stride-65 LDS tile
  {
    int r = t >> 2, seg = t & 3;
    int e = fidx[r];
    const float4* src = (const float4*)(emb + (size_t)e * 64 + seg * 16);
#pragma unroll
    for (int i = 0; i < 4; ++i) {
      float4 q4 = src[i];
      float* dst = &qlds[r * 65 + seg * 16 + i * 4];
      dst[0] = q4.x; dst[1] = q4.y; dst[2] = q4.z; dst[3] = q4.w;
    }
  }
  __syncthreads();

  // output pass: z_q_st (NHWC, same linear layout as the input chunk) + loss sums
  float ea = 0.0f, qa = 0.0f;
  const size_t ob = (size_t)bh * 4096;
#pragma unroll
  for (int j = 0; j < 16; ++j) {
    int e = t + 256 * j;                  // 0..4095 ; e = w*64 + c
    int w = e >> 6, c = e & 63;
    float v  = zb[e];                     // z value
    float q  = qlds[c * 65 + w];          // z_q value (bank-conflict-free)
    float st = q + (v - q);               // straight-through, as written in ref
    out[ob + e] = st;
    float de = q - v;  ea += de * de;     // e_latent numerator
    float dq = st - v; qa += dq * dq;     // q_latent numerator
  }
#pragma unroll
  for (int o = 16; o > 0; o >>= 1) {
    ea += __shfl_down(ea, o, 32);
    qa += __shfl_down(qa, o, 32);
  }
  if (lane == 0) {
    atomicAdd(ws + 1024, ea);
    atomicAdd(ws + 1025, qa);
  }
}

// --- epilogue: loss scalar ---------------------------------------------------
__global__ void vq_loss_kernel(const float* __restrict__ ws, float* __restrict__ out) {
  const float inv = 1.0f / (float)N_TOTAL;
  float e = ws[1024] * inv;
  float q = ws[1025] * inv;
  out[LOSS_OFF] = q + 0.25f * e;
}

extern "C" void kernel_launch(void* const* d_in, const int* in_sizes, int n_in,
                              void* d_out, int out_size, void* d_ws, size_t ws_size,
                              hipStream_t stream) {
  const float* z   = (const float*)d_in[0];   // [32,64,64,64] f32 NHWC
  const float* emb = (const float*)d_in[1];   // [1024,64] f32
  float* out = (float*)d_out;
  float* ws  = (float*)d_ws;                  // needs >= 1026 floats
  (void)in_sizes; (void)n_in; (void)out_size; (void)ws_size;

  vq_e2_kernel<<<4, 256, 0, stream>>>(emb, ws);
  vq_main_kernel<<<2048, 256, 0, stream>>>(z, emb, ws, out);
  vq_loss_kernel<<<1, 1, 0, stream>>>(ws, out);
}